// LPModel_51539608248
// MI455X (gfx1250) — compile-verified
//
#include <hip/hip_runtime.h>
#include <math.h>

#define DIMC 128
#define HID  512
#define NHEAD 8
#define HEADD 16
#define SCALE_F 0.25f   // 1/sqrt(16)

typedef __attribute__((ext_vector_type(2))) float v2f;
typedef __attribute__((ext_vector_type(8))) float v8f;
typedef __attribute__((ext_vector_type(4))) unsigned int u32x4;
typedef __attribute__((ext_vector_type(8))) int i32x8;
typedef __attribute__((ext_vector_type(4))) int i32x4;
typedef __attribute__((address_space(3))) float lds_float;

__device__ __forceinline__ unsigned ordf(float f) {
  unsigned u = __float_as_uint(f);
  return (u & 0x80000000u) ? ~u : (u | 0x80000000u);
}
__device__ __forceinline__ float unordf(unsigned u) {
  return __uint_as_float((u & 0x80000000u) ? (u & 0x7fffffffu) : ~u);
}

// ---------------------------------------------------------------------------
// Y[R x dout] = X[R x din] @ W^T + bias  (opt relu, opt residual add)
// W is [dout x din] row-major (reference: linear(x,w,b) = x @ w.T + b).
//
// Per block (256 thr = 8 waves): 16 output rows.
//  - TDM (tensor_load_to_lds) DMAs the 16 x din X-tile into LDS once;
//    OOB rows (past R) are zero-filled by the TDM.
//  - Each wave owns 16 output cols per pass; A comes from LDS (ds_load_b64),
//    B (weights) via blocked global_load_b64 (8 k-steps batched per wait).
//  - V_WMMA_F32_16X16X4_F32 accumulates in f32.
// ---------------------------------------------------------------------------
__global__ __launch_bounds__(256) void gemm_wmma_f32(
    const float* __restrict__ X, const float* __restrict__ W,
    const float* __restrict__ bias, const float* __restrict__ residual,
    float* __restrict__ Y, int R, int din, int dout, int relu)
{
  __shared__ float ldsx[16 * 512];          // max din = 512 -> 32 KB
  const int lane = threadIdx.x & 31;
  const int wave = threadIdx.x >> 5;        // 0..7
  const int row0 = blockIdx.x * 16;
  const int ntiles = dout >> 4;
  const int m  = lane & 15;                 // A row / B col inside tile
  const int kh = lane >> 4;                 // 0/1 -> which K pair this lane holds

  // ---- TDM: DMA X[row0:row0+16, 0:din] -> LDS (wave 0 issues) ----
  if (wave == 0) {
    const unsigned lds_base =
        (unsigned)(unsigned long long)(lds_float*)ldsx;       // LDS byte offset
    const unsigned long long ga =
        (unsigned long long)(const void*)(X + (size_t)row0 * din);
    const unsigned rows_left = (unsigned)((row0 < R) ? (R - row0) : 0);
    u32x4 g0;
    g0[0] = 1u;                                       // count=1, user descriptor
    g0[1] = lds_base;                                 // lds_addr
    g0[2] = (unsigned)(ga & 0xffffffffu);             // global_addr[31:0]
    g0[3] = (unsigned)((ga >> 32) & 0x01ffffffu)      // global_addr[56:32]
            | (2u << 30);                             // type = 2 ("image")
    i32x8 g1;
    g1[0] = (int)(2u << 16);                          // data_size = 4 bytes
    g1[1] = (int)(((unsigned)din & 0xffffu) << 16);   // tensor_dim0 lo16
    g1[2] = (int)(((unsigned)din >> 16) | ((rows_left & 0xffffu) << 16));
    g1[3] = (int)((rows_left >> 16) | ((unsigned)din << 16)); // tile_dim0 = din
    g1[4] = 16;                                       // tile_dim1 = 16 rows
    g1[5] = din;                                      // tensor_dim0_stride lo32
    g1[6] = 0;
    g1[7] = 0;
    i32x4 g2 = {0, 0, 0, 0};
    i32x4 g3 = {0, 0, 0, 0};
    i32x8 g4 = {0, 0, 0, 0, 0, 0, 0, 0};
    __builtin_amdgcn_tensor_load_to_lds(g0, g1, g2, g3, g4, 0);
    __builtin_amdgcn_s_wait_tensorcnt(0);
  }
  __syncthreads();

  const float* __restrict__ xl = ldsx + m * din;      // this lane's A row in LDS

  for (int nt = wave; nt < ntiles; nt += 8) {
    const int ncol = nt * 16 + m;                     // W row = output col
    const float* __restrict__ wrow = W + (size_t)ncol * din;
    v8f acc = {};
    for (int k0 = 0; k0 < din; k0 += 32) {            // 8 WMMAs per block
      v2f b[8];
#pragma unroll
      for (int t = 0; t < 8; ++t) {                   // batched B loads
        const int ka = k0 + t * 4 + kh * 2;
        b[t].x = wrow[ka];
        b[t].y = wrow[ka + 1];
      }
#pragma unroll
      for (int t = 0; t < 8; ++t) {
        const int ka = k0 + t * 4 + kh * 2;
        v2f a;
        a.x = xl[ka];                                 // ds_load_b64 from LDS
        a.y = xl[ka + 1];
        acc = __builtin_amdgcn_wmma_f32_16x16x4_f32(
                false, a, false, b[t], (short)0, acc, false, false);
      }
    }
    // D layout: VGPR j = (M=j, N=lane) lanes 0..15 ; (M=j+8, N=lane-16) 16..31
    const int dmbase = kh * 8;
    const int dn = nt * 16 + m;
    const float bv = bias ? bias[dn] : 0.0f;
#pragma unroll
    for (int j = 0; j < 8; ++j) {
      const int orow = row0 + dmbase + j;
      if (orow < R) {
        float v = acc[j] + bv;
        if (relu) v = fmaxf(v, 0.0f);
        if (residual) v += residual[(size_t)orow * dout + dn];
        Y[(size_t)orow * dout + dn] = v;
      }
    }
  }
}

// ---------------------------------------------------------------------------
// LayerNorm over last dim (=128). One wave per row, float4 per lane.
// ---------------------------------------------------------------------------
__global__ __launch_bounds__(256) void layernorm_k(
    const float* __restrict__ X, const float* __restrict__ g,
    const float* __restrict__ b, float* __restrict__ Y, int R)
{
  const int lane = threadIdx.x & 31;
  const int row = blockIdx.x * 8 + (threadIdx.x >> 5);
  if (row >= R) return;
  float4 v = ((const float4*)(X + (size_t)row * DIMC))[lane];
  float s = v.x + v.y + v.z + v.w;
#pragma unroll
  for (int o = 16; o; o >>= 1) s += __shfl_xor(s, o, 32);
  const float mean = s * (1.0f / DIMC);
  const float dx = v.x - mean, dy = v.y - mean, dz = v.z - mean, dw = v.w - mean;
  float q = dx * dx + dy * dy + dz * dz + dw * dw;
#pragma unroll
  for (int o = 16; o; o >>= 1) q += __shfl_xor(q, o, 32);
  const float inv = rsqrtf(q * (1.0f / DIMC) + 1e-5f);
  const float4 gg = ((const float4*)g)[lane];
  const float4 bb = ((const float4*)b)[lane];
  float4 o;
  o.x = dx * inv * gg.x + bb.x;
  o.y = dy * inv * gg.y + bb.y;
  o.z = dz * inv * gg.z + bb.z;
  o.w = dw * inv * gg.w + bb.w;
  ((float4*)(Y + (size_t)row * DIMC))[lane] = o;
}

// ---------------------------------------------------------------------------
// Node attention pass 1: score per (edge, head) + segment max over src.
// Q = eQ[e] + nQ[src], K = eK[e] + nK[dst]
// ---------------------------------------------------------------------------
__global__ __launch_bounds__(256) void node_score_max(
    const float* __restrict__ eQ, const float* __restrict__ eK,
    const float* __restrict__ nQ, const float* __restrict__ nK,
    const int* __restrict__ ei, int M, float* __restrict__ sbuf,
    unsigned* __restrict__ mseg)
{
  const int t = blockIdx.x * blockDim.x + threadIdx.x;
  if (t >= M * NHEAD) return;
  const int e = t >> 3, h = t & 7;
  const int src = ei[e], dst = ei[M + e];
  const float* q1 = eQ + (size_t)e * DIMC + h * HEADD;
  const float* q2 = nQ + (size_t)src * DIMC + h * HEADD;
  const float* k1 = eK + (size_t)e * DIMC + h * HEADD;
  const float* k2 = nK + (size_t)dst * DIMC + h * HEADD;
  float s = 0.0f;
#pragma unroll
  for (int i = 0; i < HEADD; ++i) s += (q1[i] + q2[i]) * (k1[i] + k2[i]);
  s *= SCALE_F;
  sbuf[t] = s;
  atomicMax(mseg + (size_t)src * NHEAD + h, ordf(s));
}

// Shared pass 2: ex = exp(s - m[seg]); den[seg] += ex.  seg = idx[e*stride].
__global__ __launch_bounds__(256) void seg_expsum(
    float* __restrict__ sbuf, const unsigned* __restrict__ mseg,
    float* __restrict__ den, const int* __restrict__ idx, int stride, int count)
{
  const int t = blockIdx.x * blockDim.x + threadIdx.x;
  if (t >= count * NHEAD) return;
  const int e = t >> 3, h = t & 7;
  const int s0 = idx[(size_t)e * stride];
  const float m = unordf(mseg[(size_t)s0 * NHEAD + h]);
  const float ex = __expf(sbuf[t] - m);
  sbuf[t] = ex;
  atomicAdd(den + (size_t)s0 * NHEAD + h, ex);
}

// Node attention pass 3: agg[src] += (ex/den) * (eV[e] + nV[dst])
__global__ __launch_bounds__(256) void node_scatter(
    const float* __restrict__ sbuf, const float* __restrict__ den,
    const float* __restrict__ eV, const float* __restrict__ nV,
    const int* __restrict__ ei, int M, float* __restrict__ agg)
{
  const size_t t = (size_t)blockIdx.x * blockDim.x + threadIdx.x;
  if (t >= (size_t)M * DIMC) return;
  const int e = (int)(t >> 7), d = (int)(t & 127), h = d >> 4;
  const int src = ei[e], dst = ei[M + e];
  const float w = sbuf[(size_t)e * NHEAD + h] / den[(size_t)src * NHEAD + h];
  const float v = eV[(size_t)e * DIMC + d] + nV[(size_t)dst * DIMC + d];
  atomicAdd(agg + (size_t)src * DIMC + d, w * v);
}

// Fold node terms into per-edge Q/K/V in place:  eX[e] += nX[src] + nX[dst]
__global__ __launch_bounds__(256) void edge_combine(
    float* __restrict__ eQ, float* __restrict__ eK, float* __restrict__ eV,
    const float* __restrict__ nQ, const float* __restrict__ nK,
    const float* __restrict__ nV, const int* __restrict__ ei, int M)
{
  const size_t t = (size_t)blockIdx.x * blockDim.x + threadIdx.x;
  if (t >= (size_t)M * DIMC) return;
  const int e = (int)(t >> 7), d = (int)(t & 127);
  const int src = ei[e], dst = ei[M + e];
  const size_t o  = (size_t)e * DIMC + d;
  const size_t s1 = (size_t)src * DIMC + d;
  const size_t s2 = (size_t)dst * DIMC + d;
  eQ[o] += nQ[s1] + nQ[s2];
  eK[o] += nK[s1] + nK[s2];
  eV[o] += nV[s1] + nV[s2];
}

// Pair attention pass 1: score per (pair, head) + segment max over a.
__global__ __launch_bounds__(256) void pair_score_max(
    const float* __restrict__ Qe, const float* __restrict__ Ke,
    const int* __restrict__ pairs, int P, float* __restrict__ sbuf,
    unsigned* __restrict__ mseg)
{
  const int t = blockIdx.x * blockDim.x + threadIdx.x;
  if (t >= P * NHEAD) return;
  const int p = t >> 3, h = t & 7;
  const int a = pairs[(size_t)p * 2], b = pairs[(size_t)p * 2 + 1];
  const float* q = Qe + (size_t)a * DIMC + h * HEADD;
  const float* k = Ke + (size_t)b * DIMC + h * HEADD;
  float s = 0.0f;
#pragma unroll
  for (int i = 0; i < HEADD; ++i) s += q[i] * k[i];
  s *= SCALE_F;
  sbuf[t] = s;
  atomicMax(mseg + (size_t)a * NHEAD + h, ordf(s));
}

// Pair attention pass 3: agg[a] += (ex/den) * Ve[b]
__global__ __launch_bounds__(256) void pair_scatter(
    const float* __restrict__ sbuf, const float* __restrict__ den,
    const float* __restrict__ Ve, const int* __restrict__ pairs, int P,
    float* __restrict__ agg)
{
  const size_t t = (size_t)blockIdx.x * blockDim.x + threadIdx.x;
  if (t >= (size_t)P * DIMC) return;
  const int p = (int)(t >> 7), d = (int)(t & 127), h = d >> 4;
  const int a = pairs[(size_t)p * 2], b = pairs[(size_t)p * 2 + 1];
  const float w = sbuf[(size_t)p * NHEAD + h] / den[(size_t)a * NHEAD + h];
  atomicAdd(agg + (size_t)a * DIMC + d, w * Ve[(size_t)b * DIMC + d]);
}

// ---------------------------------------------------------------------------
extern "C" void kernel_launch(void* const* d_in, const int* in_sizes, int n_in,
                              void* d_out, int out_size, void* d_ws, size_t ws_size,
                              hipStream_t stream)
{
  const float* node_t = (const float*)d_in[0];
  const float* edge_t = (const float*)d_in[1];
  const int*   ei     = (const int*)d_in[2];
  const int*   pairs  = (const int*)d_in[3];
  const int N = in_sizes[0] / DIMC;
  const int M = in_sizes[1] / DIMC;
  const int P = in_sizes[3] / 2;

  // params: jax pytree leaf order = nested dict keys sorted alphabetically
  const float* e_fc1_b   = (const float*)d_in[4];
  const float* e_fc1_w   = (const float*)d_in[5];
  const float* e_fc2_b   = (const float*)d_in[6];
  const float* e_fc2_w   = (const float*)d_in[7];
  const float* e_ffn_lnb = (const float*)d_in[8];
  const float* e_ffn_lng = (const float*)d_in[9];
  const float* e_ln1_b   = (const float*)d_in[10];
  const float* e_ln1_g   = (const float*)d_in[11];
  const float* e_ln2_b   = (const float*)d_in[12];
  const float* e_ln2_g   = (const float*)d_in[13];
  const float* em_Wek_b = (const float*)d_in[14];
  const float* em_Wek_w = (const float*)d_in[15];
  const float* em_Weq_b = (const float*)d_in[16];
  const float* em_Weq_w = (const float*)d_in[17];
  const float* em_Wev_b = (const float*)d_in[18];
  const float* em_Wev_w = (const float*)d_in[19];
  const float* em_Wnk_b = (const float*)d_in[20];
  const float* em_Wnk_w = (const float*)d_in[21];
  const float* em_Wnq_b = (const float*)d_in[22];
  const float* em_Wnq_w = (const float*)d_in[23];
  const float* em_Wnv_b = (const float*)d_in[24];
  const float* em_Wnv_w = (const float*)d_in[25];
  const float* em_out_b = (const float*)d_in[26];
  const float* em_out_w = (const float*)d_in[27];
  const float* n_fc1_b   = (const float*)d_in[28];
  const float* n_fc1_w   = (const float*)d_in[29];
  const float* n_fc2_b   = (const float*)d_in[30];
  const float* n_fc2_w   = (const float*)d_in[31];
  const float* n_ffn_lnb = (const float*)d_in[32];
  const float* n_ffn_lng = (const float*)d_in[33];
  const float* n_ln1_b   = (const float*)d_in[34];
  const float* n_ln1_g   = (const float*)d_in[35];
  const float* n_ln2_b   = (const float*)d_in[36];
  const float* n_ln2_g   = (const float*)d_in[37];
  const float* nm_Wek_b = (const float*)d_in[38];
  const float* nm_Wek_w = (const float*)d_in[39];
  const float* nm_Weq_b = (const float*)d_in[40];
  const float* nm_Weq_w = (const float*)d_in[41];
  const float* nm_Wev_b = (const float*)d_in[42];
  const float* nm_Wev_w = (const float*)d_in[43];
  const float* nm_Wnk_b = (const float*)d_in[44];
  const float* nm_Wnk_w = (const float*)d_in[45];
  const float* nm_Wnq_b = (const float*)d_in[46];
  const float* nm_Wnq_w = (const float*)d_in[47];
  const float* nm_Wnv_b = (const float*)d_in[48];
  const float* nm_Wnv_w = (const float*)d_in[49];
  const float* nm_out_b = (const float*)d_in[50];
  const float* nm_out_w = (const float*)d_in[51];

  float* n_new = (float*)d_out;                 // [N,128]
  float* e_new = n_new + (size_t)N * DIMC;      // [M,128]

  // ---- workspace layout ----
  const int RM = (M > N) ? M : N;
  float* ws = (float*)d_ws;
  size_t off = 0;
  auto alloc = [&](size_t nelem) { float* p = ws + off; off += nelem; return p; };
  float* n_ln   = alloc((size_t)N * DIMC);
  float* e_ln   = alloc((size_t)M * DIMC);
  float* nQ     = alloc((size_t)N * DIMC);
  float* nK     = alloc((size_t)N * DIMC);
  float* nV     = alloc((size_t)N * DIMC);
  float* eQ     = alloc((size_t)M * DIMC);
  float* eK     = alloc((size_t)M * DIMC);
  float* eV     = alloc((size_t)M * DIMC);
  float* exE    = alloc((size_t)M * NHEAD);
  float* mN     = alloc((size_t)N * NHEAD);     // used as unsigned
  float* denN   = alloc((size_t)N * NHEAD);
  float* aggN   = alloc((size_t)N * DIMC);
  float* nprime = alloc((size_t)N * DIMC);
  float* lntmp  = alloc((size_t)RM * DIMC);
  float* hbuf   = alloc((size_t)RM * HID);      // FFN hidden; aliased as pair ex
  float* mE     = alloc((size_t)M * NHEAD);
  float* denE   = alloc((size_t)M * NHEAD);
  float* aggE   = alloc((size_t)M * DIMC);
  float* eprime = alloc((size_t)M * DIMC);
  float* exP    = hbuf;                         // pair scores: idle window of hbuf

  auto cdiv = [](long a, long b) { return (unsigned)((a + b - 1) / b); };
  const unsigned gN16 = cdiv(N, 16), gM16 = cdiv(M, 16);
  const unsigned gNr = cdiv(N, 8), gMr = cdiv(M, 8);
  const unsigned gMH = cdiv((long)M * NHEAD, 256);
  const unsigned gPH = cdiv((long)P * NHEAD, 256);
  const unsigned gMD = cdiv((long)M * DIMC, 256);
  const unsigned gPD = cdiv((long)P * DIMC, 256);

  // ================= Node phase =================
  layernorm_k<<<gNr, 256, 0, stream>>>(node_t, n_ln1_g, n_ln1_b, n_ln, N);
  layernorm_k<<<gMr, 256, 0, stream>>>(edge_t, e_ln1_g, e_ln1_b, e_ln, M);

  gemm_wmma_f32<<<gN16, 256, 0, stream>>>(n_ln, nm_Wnq_w, nm_Wnq_b, nullptr, nQ, N, DIMC, DIMC, 0);
  gemm_wmma_f32<<<gN16, 256, 0, stream>>>(n_ln, nm_Wnk_w, nm_Wnk_b, nullptr, nK, N, DIMC, DIMC, 0);
  gemm_wmma_f32<<<gN16, 256, 0, stream>>>(n_ln, nm_Wnv_w, nm_Wnv_b, nullptr, nV, N, DIMC, DIMC, 0);
  gemm_wmma_f32<<<gM16, 256, 0, stream>>>(e_ln, nm_Weq_w, nm_Weq_b, nullptr, eQ, M, DIMC, DIMC, 0);
  gemm_wmma_f32<<<gM16, 256, 0, stream>>>(e_ln, nm_Wek_w, nm_Wek_b, nullptr, eK, M, DIMC, DIMC, 0);
  gemm_wmma_f32<<<gM16, 256, 0, stream>>>(e_ln, nm_Wev_w, nm_Wev_b, nullptr, eV, M, DIMC, DIMC, 0);

  (void)hipMemsetAsync(mN, 0, (size_t)N * NHEAD * 4, stream);
  (void)hipMemsetAsync(denN, 0, (size_t)N * NHEAD * 4, stream);
  (void)hipMemsetAsync(aggN, 0, (size_t)N * DIMC * 4, stream);

  node_score_max<<<gMH, 256, 0, stream>>>(eQ, eK, nQ, nK, ei, M, exE, (unsigned*)mN);
  seg_expsum<<<gMH, 256, 0, stream>>>(exE, (const unsigned*)mN, denN, ei, 1, M);
  node_scatter<<<gMD, 256, 0, stream>>>(exE, denN, eV, nV, ei, M, aggN);

  gemm_wmma_f32<<<gN16, 256, 0, stream>>>(aggN, nm_out_w, nm_out_b, node_t, nprime, N, DIMC, DIMC, 0);

  // node FFN: n_new = fc2(relu(fc1(LN(nprime)))) + nprime
  layernorm_k<<<gNr, 256, 0, stream>>>(nprime, n_ffn_lng, n_ffn_lnb, lntmp, N);
  gemm_wmma_f32<<<gN16, 256, 0, stream>>>(lntmp, n_fc1_w, n_fc1_b, nullptr, hbuf, N, DIMC, HID, 1);
  gemm_wmma_f32<<<gN16, 256, 0, stream>>>(hbuf, n_fc2_w, n_fc2_b, nprime, n_new, N, HID, DIMC, 0);

  // ================= Edge phase =================
  layernorm_k<<<gNr, 256, 0, stream>>>(n_new, n_ln2_g, n_ln2_b, n_ln, N);
  layernorm_k<<<gMr, 256, 0, stream>>>(edge_t, e_ln2_g, e_ln2_b, e_ln, M);

  gemm_wmma_f32<<<gN16, 256, 0, stream>>>(n_ln, em_Wnq_w, em_Wnq_b, nullptr, nQ, N, DIMC, DIMC, 0);
  gemm_wmma_f32<<<gN16, 256, 0, stream>>>(n_ln, em_Wnk_w, em_Wnk_b, nullptr, nK, N, DIMC, DIMC, 0);
  gemm_wmma_f32<<<gN16, 256, 0, stream>>>(n_ln, em_Wnv_w, em_Wnv_b, nullptr, nV, N, DIMC, DIMC, 0);
  gemm_wmma_f32<<<gM16, 256, 0, stream>>>(e_ln, em_Weq_w, em_Weq_b, nullptr, eQ, M, DIMC, DIMC, 0);
  gemm_wmma_f32<<<gM16, 256, 0, stream>>>(e_ln, em_Wek_w, em_Wek_b, nullptr, eK, M, DIMC, DIMC, 0);
  gemm_wmma_f32<<<gM16, 256, 0, stream>>>(e_ln, em_Wev_w, em_Wev_b, nullptr, eV, M, DIMC, DIMC, 0);

  // eQ/eK/eV become per-edge Qe/Ke/Ve (add both endpoint node terms)
  edge_combine<<<gMD, 256, 0, stream>>>(eQ, eK, eV, nQ, nK, nV, ei, M);

  (void)hipMemsetAsync(mE, 0, (size_t)M * NHEAD * 4, stream);
  (void)hipMemsetAsync(denE, 0, (size_t)M * NHEAD * 4, stream);
  (void)hipMemsetAsync(aggE, 0, (size_t)M * DIMC * 4, stream);

  if (P > 0) {
    pair_score_max<<<gPH, 256, 0, stream>>>(eQ, eK, pairs, P, exP, (unsigned*)mE);
    seg_expsum<<<gPH, 256, 0, stream>>>(exP, (const unsigned*)mE, denE, pairs, 2, P);
    pair_scatter<<<gPD, 256, 0, stream>>>(exP, denE, eV, pairs, P, aggE);
  }

  gemm_wmma_f32<<<gM16, 256, 0, stream>>>(aggE, em_out_w, em_out_b, edge_t, eprime, M, DIMC, DIMC, 0);

  // edge FFN: e_new = fc2(relu(fc1(LN(eprime)))) + eprime
  layernorm_k<<<gMr, 256, 0, stream>>>(eprime, e_ffn_lng, e_ffn_lnb, lntmp, M);
  gemm_wmma_f32<<<gM16, 256, 0, stream>>>(lntmp, e_fc1_w, e_fc1_b, nullptr, hbuf, M, DIMC, HID, 1);
  gemm_wmma_f32<<<gM16, 256, 0, stream>>>(hbuf, e_fc2_w, e_fc2_b, eprime, e_new, M, HID, DIMC, 0);
}